// User_91345364451436
// MI455X (gfx1250) — compile-verified
//
#include <hip/hip_runtime.h>
#include <hip/hip_bf16.h>

// ---------------------------------------------------------------------------
// Problem: out = relu(segment_sum(val * (embs@Wp)[col], row) + user@Wu + bias)
// Optimization: S @ (embs @ Wp) == (S @ embs) @ Wp  (matmul associativity)
//   Phase 1: agg[row,:] += val * embs[col,:]        (scatter, L2-resident)
//   Phase 2: out = relu(agg @ Wp + user @ Wu + b)   (fp32 WMMA, fused)
// ---------------------------------------------------------------------------

typedef __attribute__((ext_vector_type(2))) float v2f;
typedef __attribute__((ext_vector_type(8))) float v8f;

#define DIM 256

// ---------------- Phase 0: zero the segment-sum accumulator -----------------
__global__ void zero_kernel(float4* __restrict__ p, int n4) {
    int i = blockIdx.x * blockDim.x + threadIdx.x;
    if (i < n4) p[i] = make_float4(0.f, 0.f, 0.f, 0.f);
}

// ---------------- Phase 1: edge scatter-add of raw embeddings ---------------
// 4 edges per 256-thread block; 64 lanes per edge, each lane moves a float4.
// Source rows (102 MB) and destination (16.8 MB) are both L2-resident on the
// 192 MB global L2, so atomics resolve at L2 rather than HBM.
__global__ __launch_bounds__(256)
void scatter_kernel(const float* __restrict__ embs,
                    const int*   __restrict__ erow,
                    const int*   __restrict__ ecol,
                    const float* __restrict__ eval,
                    float*       __restrict__ agg,
                    int nEdges) {
    const int sub = threadIdx.x >> 6;   // edge within block (0..3)
    const int t   = threadIdx.x & 63;   // float4 chunk within row
    const int e   = blockIdx.x * 4 + sub;
    if (e >= nEdges) return;

    const int   row = erow[e];
    const int   col = ecol[e];
    const float v   = eval[e];

    const float4 x = ((const float4*)(embs + (size_t)col * DIM))[t];
    float* dst = agg + (size_t)row * DIM + t * 4;
    atomicAdd(dst + 0, v * x.x);
    atomicAdd(dst + 1, v * x.y);
    atomicAdd(dst + 2, v * x.z);
    atomicAdd(dst + 3, v * x.w);
}

// ---------------- Phase 2: fused dual GEMM + bias + relu --------------------
// out[16384,256] = relu(agg@Wp + user@Wu + bias), fp32 WMMA 16x16x4.
// Block = 128 threads (4 waves) covers a 16-row x 256-col output tile.
// Each wave owns a 16x64 strip: 4 v8f accumulators, A-frag reused 4x.
__global__ __launch_bounds__(128)
void fused_gemm_kernel(const float* __restrict__ agg,   // [M, DIM]
                       const float* __restrict__ user,  // [M, DIM]
                       const float* __restrict__ Wp,    // [DIM, DIM]
                       const float* __restrict__ Wu,    // [DIM, DIM]
                       const float* __restrict__ bias,  // [DIM]
                       float*       __restrict__ out,   // [M, DIM]
                       int M) {
    constexpr int K = DIM, N = DIM;
    const int wave = threadIdx.x >> 5;          // 0..3 -> 64-col strip
    const int lane = threadIdx.x & 31;
    const int half = lane >> 4;                 // 0: lanes 0-15, 1: lanes 16-31
    const int l16  = lane & 15;
    const int row0 = blockIdx.x * 16;
    const int col0 = wave * 64;

    v8f acc[4];
    #pragma unroll
    for (int j = 0; j < 4; ++j) acc[j] = v8f{};

    const float* aggRow = agg  + (size_t)(row0 + l16) * K;
    const float* usrRow = user + (size_t)(row0 + l16) * K;

    for (int k0 = 0; k0 < K; k0 += 4) {
        // A fragment, 16x4 f32 layout (ISA 7.12.2):
        //   lanes 0-15: VGPR0=K0, VGPR1=K1 ; lanes 16-31: VGPR0=K2, VGPR1=K3
        const int ka = k0 + 2 * half;
        v2f aA, aU;
        aA.x = aggRow[ka];  aA.y = aggRow[ka + 1];
        aU.x = usrRow[ka];  aU.y = usrRow[ka + 1];

        #pragma unroll
        for (int j = 0; j < 4; ++j) {
            // B fragment, 4x16 f32: rows (K) striped across VGPRs, lane = N,
            // lanes 16-31 carry K=2,3 (mirror of A layout).
            const int c = col0 + 16 * j + l16;
            v2f bP, bU;
            bP.x = Wp[(size_t)ka * N + c];
            bP.y = Wp[(size_t)(ka + 1) * N + c];
            bU.x = Wu[(size_t)ka * N + c];
            bU.y = Wu[(size_t)(ka + 1) * N + c];

            acc[j] = __builtin_amdgcn_wmma_f32_16x16x4_f32(
                false, aA, false, bP, (short)0, acc[j], false, false);
            acc[j] = __builtin_amdgcn_wmma_f32_16x16x4_f32(
                false, aU, false, bU, (short)0, acc[j], false, false);
        }
    }

    // Epilogue: C/D layout — VGPR v holds M = v + 8*half, N = l16.
    #pragma unroll
    for (int j = 0; j < 4; ++j) {
        const int c = col0 + 16 * j + l16;
        const float b = bias[c];
        #pragma unroll
        for (int v = 0; v < 8; ++v) {
            const int r = row0 + v + 8 * half;
            float val = acc[j][v] + b;
            out[(size_t)r * N + c] = val > 0.f ? val : 0.f;
        }
    }
}

// ---------------------------------------------------------------------------
extern "C" void kernel_launch(void* const* d_in, const int* in_sizes, int n_in,
                              void* d_out, int out_size, void* d_ws, size_t ws_size,
                              hipStream_t stream) {
    const float* embs = (const float*)d_in[0];  // [100000, 256]
    const float* user = (const float*)d_in[1];  // [16384, 256]
    const int*   erow = (const int*)  d_in[2];  // [n_edges]
    const int*   ecol = (const int*)  d_in[3];  // [n_edges]
    const float* eval = (const float*)d_in[4];  // [n_edges]
    const float* Wp   = (const float*)d_in[5];  // [256, 256]
    const float* Wu   = (const float*)d_in[6];  // [256, 256]
    const float* bias = (const float*)d_in[7];  // [256]
    float*       out  = (float*)d_out;

    const int M      = in_sizes[1] / DIM;       // 16384
    const int nEdges = in_sizes[2];             // 1048576

    float* agg = (float*)d_ws;                  // [M, DIM] = 16.8 MB scratch

    const int n4 = M * DIM / 4;
    zero_kernel<<<(n4 + 255) / 256, 256, 0, stream>>>((float4*)agg, n4);

    scatter_kernel<<<(nEdges + 3) / 4, 256, 0, stream>>>(
        embs, erow, ecol, eval, agg, nEdges);

    fused_gemm_kernel<<<M / 16, 128, 0, stream>>>(
        agg, user, Wp, Wu, bias, out, M);
}